// HBV_37658273251764
// MI455X (gfx1250) — compile-verified
//
#include <hip/hip_runtime.h>

// HBV hydrological model on gfx1250 (MI455X): strictly sequential nonlinear
// scan, 365 steps x 50000 independent cells. Issue/VALU bound (~70 VALU + 2
// TRANS per cell-step vs a ~12.5us HBM floor), so:
//   * 2 independent cells per thread -> two interleaved dependency chains in
//     one instruction stream (VOPD dual-issue + VALU/TRANS co-execution)
//   * states + denormalized params live in VGPRs for the whole scan
//   * branchless, clamp-free loop body; last step peeled
//   * coalesced NT b96 loads / NT b32 stores (read-once / write-once streams)
//   * speculative global_prefetch_b8 8 steps (~4.8 MB) ahead (never faults)

#define HBV_NSTEP 365
#define HBV_NGRID 50000
#define HBV_HALF  (HBV_NGRID / 2)
#define HBV_PRECS 1e-5f

struct HbvCell {
    // denormalized parameters (+ derived constants)
    float BETA, FC, K0, K1, K2, PERCp, UZL, TT, CFMAX, CWH;
    float invFC, invLPFC, refrC;
    // state
    float SP, MW, SM, SUZ, SLZ;
};

__device__ __forceinline__ void hbv_init(const float* __restrict__ params,
                                         int g, HbvCell& c) {
    const float lo[12] = {1.0f, 50.0f,  0.05f, 0.01f, 0.001f, 0.2f,
                          0.0f, 0.0f,  -2.5f,  0.5f,  0.0f,   0.0f};
    const float hi[12] = {6.0f, 1000.0f, 0.9f, 0.5f,  0.2f,   1.0f,
                          10.0f, 100.0f, 2.5f, 10.0f, 0.1f,   0.2f};
    const float* pg = params + (size_t)g * 14;
    float p[12];
#pragma unroll
    for (int i = 0; i < 12; ++i) {
        p[i] = __builtin_fmaf(pg[i], hi[i] - lo[i], lo[i]);
    }
    c.BETA = p[0]; c.FC = p[1]; c.K0 = p[2]; c.K1 = p[3]; c.K2 = p[4];
    const float LP = p[5];
    c.PERCp = p[6]; c.UZL = p[7]; c.TT = p[8]; c.CFMAX = p[9];
    const float CFR = p[10];
    c.CWH = p[11];
    c.invFC   = 1.0f / c.FC;
    c.invLPFC = 1.0f / (LP * c.FC);
    c.refrC   = CFR * c.CFMAX;
    c.SP = c.MW = c.SM = c.SUZ = c.SLZ = 0.001f;
}

__device__ __forceinline__ float hbv_step(HbvCell& c, float P, float T, float E) {
    // snow routine
    const float RAIN = (T >= c.TT) ? P : 0.0f;          // P * is_rain
    const float SNOW = P - RAIN;                        // P * (1 - is_rain)
    c.SP += SNOW;
    const float melt = fminf(fmaxf(c.CFMAX * (T - c.TT), 0.0f), c.SP);
    c.MW += melt;
    c.SP -= melt;
    const float refr = fminf(fmaxf(c.refrC * (c.TT - T), 0.0f), c.MW);
    c.SP += refr;
    c.MW -= refr;
    const float tosoil = fmaxf(c.MW - c.CWH * c.SP, 0.0f);
    c.MW -= tosoil;

    // soil routine: (SM/FC)^beta via v_log_f32/v_exp_f32 (SM in [PRECS, FC])
    const float ratio = c.SM * c.invFC;
    float sw = __builtin_amdgcn_exp2f(c.BETA * __builtin_amdgcn_logf(ratio));
    sw = fminf(fmaxf(sw, 0.0f), 1.0f);                  // -> v_med3-style clamp
    const float recharge = (RAIN + tosoil) * sw;
    c.SM = c.SM + RAIN + tosoil - recharge;
    const float excess = fmaxf(c.SM - c.FC, 0.0f);
    c.SM -= excess;
    const float evap  = fminf(fmaxf(c.SM * c.invLPFC, 0.0f), 1.0f);
    const float ETact = fminf(c.SM, E * evap);
    c.SM = fmaxf(c.SM - ETact, HBV_PRECS);

    // response routine
    c.SUZ += recharge + excess;
    const float PERC = fminf(c.SUZ, c.PERCp);
    c.SUZ -= PERC;
    const float Q0 = c.K0 * fmaxf(c.SUZ - c.UZL, 0.0f);
    c.SUZ -= Q0;
    const float Q1 = c.K1 * c.SUZ;
    c.SUZ -= Q1;
    c.SLZ += PERC;
    const float Q2 = c.K2 * c.SLZ;
    c.SLZ -= Q2;
    return Q0 + Q1 + Q2;
}

__global__ __launch_bounds__(256)
void hbv_forward_kernel(const float* __restrict__ x,
                        const float* __restrict__ params,
                        float* __restrict__ out) {
    const int i = blockIdx.x * blockDim.x + threadIdx.x;
    if (i >= HBV_HALF) return;

    const int g0 = i;               // two independent cells per thread:
    const int g1 = i + HBV_HALF;    // both streams stay lane-coalesced

    HbvCell c0, c1;
    hbv_init(params, g0, c0);
    hbv_init(params, g1, c1);

    const float* x0 = x + (size_t)g0 * 3;   // (t, g, c): step stride = 3*NGRID
    const float* x1 = x + (size_t)g1 * 3;

    // software pipeline prologue: step 0's forcing into registers
    float P0 = __builtin_nontemporal_load(x0 + 0);
    float T0 = __builtin_nontemporal_load(x0 + 1);
    float E0 = __builtin_nontemporal_load(x0 + 2);
    float P1 = __builtin_nontemporal_load(x1 + 0);
    float T1 = __builtin_nontemporal_load(x1 + 1);
    float E1 = __builtin_nontemporal_load(x1 + 2);

    // steps 0..363: always safe to touch t+1; body is branchless & clamp-free
    for (int t = 0; t < HBV_NSTEP - 1; ++t) {
        // speculative prefetch (global_prefetch_b8) ~8 steps ahead; a
        // speculative prefetch never faults, so no end-of-buffer clamp needed
        __builtin_prefetch(x0 + (size_t)(t + 8) * (3 * HBV_NGRID), 0, 0);
        __builtin_prefetch(x1 + (size_t)(t + 8) * (3 * HBV_NGRID), 0, 0);

        // issue next step's loads before the dependent compute chains
        const float* n0 = x0 + (size_t)(t + 1) * (3 * HBV_NGRID);
        const float* n1 = x1 + (size_t)(t + 1) * (3 * HBV_NGRID);
        const float P0n = __builtin_nontemporal_load(n0 + 0);
        const float T0n = __builtin_nontemporal_load(n0 + 1);
        const float E0n = __builtin_nontemporal_load(n0 + 2);
        const float P1n = __builtin_nontemporal_load(n1 + 0);
        const float T1n = __builtin_nontemporal_load(n1 + 1);
        const float E1n = __builtin_nontemporal_load(n1 + 2);

        // two independent dependency chains -> VOPD dual-issue / TRANS co-exec
        const float Qa = hbv_step(c0, P0, T0, E0);
        const float Qb = hbv_step(c1, P1, T1, E1);

        __builtin_nontemporal_store(Qa, out + (size_t)t * HBV_NGRID + g0);
        __builtin_nontemporal_store(Qb, out + (size_t)t * HBV_NGRID + g1);

        P0 = P0n; T0 = T0n; E0 = E0n;
        P1 = P1n; T1 = T1n; E1 = E1n;
    }

    // peeled final step (t = NSTEP-1): no next-step load
    {
        const int t = HBV_NSTEP - 1;
        const float Qa = hbv_step(c0, P0, T0, E0);
        const float Qb = hbv_step(c1, P1, T1, E1);
        __builtin_nontemporal_store(Qa, out + (size_t)t * HBV_NGRID + g0);
        __builtin_nontemporal_store(Qb, out + (size_t)t * HBV_NGRID + g1);
    }
}

extern "C" void kernel_launch(void* const* d_in, const int* in_sizes, int n_in,
                              void* d_out, int out_size, void* d_ws, size_t ws_size,
                              hipStream_t stream) {
    (void)in_sizes; (void)n_in; (void)out_size; (void)d_ws; (void)ws_size;
    const float* x      = (const float*)d_in[0];   // (NSTEP, NGRID, 3) fp32
    const float* params = (const float*)d_in[1];   // (NGRID, 14)       fp32
    float*       out    = (float*)d_out;           // (NSTEP, NGRID, 1) fp32

    const int block = 256;                          // 8 wave32s per workgroup
    const int grid  = (HBV_HALF + block - 1) / block;
    hbv_forward_kernel<<<grid, block, 0, stream>>>(x, params, out);
}